// LFMMILoss_52561809768629
// MI455X (gfx1250) — compile-verified
//
#include <hip/hip_runtime.h>

#define NEG_INF   (-1e30f)
#define BATCH     16
#define TMAX      500
#define NPDF      3000
#define S_DEN     2000
#define A_DEN     48000
#define S_NUM     251
#define A_NUM     501
#define NTHR      1024

// ---------- ordered-uint encoding so float max can use ds_max_u32 ----------
__device__ __forceinline__ unsigned ford(float f) {
  unsigned u = __float_as_uint(f);
  return (u & 0x80000000u) ? ~u : (u | 0x80000000u);
}
__device__ __forceinline__ float funord(unsigned u) {
  return (u & 0x80000000u) ? __uint_as_float(u ^ 0x80000000u)
                           : __uint_as_float(~u);
}

// ---------- async global -> LDS staging of one ll row (12000 B, b128) ------
__device__ __forceinline__ void stage_row(const float* __restrict__ g,
                                          float* l, int nfloats,
                                          int tid, int nthr) {
  const int nchunk = nfloats >> 2;  // 16-byte chunks
#if __has_builtin(__builtin_amdgcn_global_load_async_to_lds_b128)
  typedef int v4i __attribute__((ext_vector_type(4)));
  typedef __attribute__((address_space(1))) v4i g_v4i;
  typedef __attribute__((address_space(3))) v4i l_v4i;
  for (int i = tid; i < nchunk; i += nthr) {
    g_v4i* gp = (g_v4i*)(unsigned long long)(const void*)(g + 4 * i);
    l_v4i* lp = (l_v4i*)(unsigned)(unsigned long long)(const void*)(l + 4 * i);
    __builtin_amdgcn_global_load_async_to_lds_b128(gp, lp, 0, 0);
  }
#else
  const float4* g4 = (const float4*)g;
  float4* l4 = (float4*)l;
  for (int i = tid; i < nchunk; i += nthr) l4[i] = g4[i];
#endif
}

__device__ __forceinline__ void wait_async0() {
#if __has_builtin(__builtin_amdgcn_s_wait_asynccnt)
  __builtin_amdgcn_s_wait_asynccnt(0);
#elif defined(__gfx1250__)
  asm volatile("s_wait_asynccnt 0" ::: "memory");
#endif
}

// ---------- fused forward (alpha) recursion: blocks 0..15 den, 16..31 num ---
__global__ __launch_bounds__(NTHR) void lfmmi_alpha_kernel(
    const float* __restrict__ ll,       // [B,T,C]
    const int* __restrict__ seqlen,     // [B]
    const int* __restrict__ nsrc, const int* __restrict__ ndst,
    const int* __restrict__ npdf, const float* __restrict__ nw,
    const float* __restrict__ ninit, const float* __restrict__ nfin,
    const int* __restrict__ dsrc, const int* __restrict__ ddst,
    const int* __restrict__ dpdf, const float* __restrict__ dw,
    const float* __restrict__ dinit, const float* __restrict__ dfin,
    float* __restrict__ llh_out)        // [32]: 0..15 den, 16..31 num
{
  __shared__ float    s_alpha[2048];
  __shared__ unsigned s_m[2048];
  __shared__ float    s_sum[2048];
  __shared__ float    s_ll[2][3008];
  extern __shared__ __align__(16) float s_score[];   // A_DEN floats (dynamic)

  const int tid    = threadIdx.x;
  const bool isDen = (blockIdx.x < BATCH);
  const int b      = blockIdx.x & (BATCH - 1);

  const int S = isDen ? S_DEN : S_NUM;
  const int* __restrict__ src  = isDen ? dsrc : (nsrc + b * A_NUM);
  const int* __restrict__ dst  = isDen ? ddst : (ndst + b * A_NUM);
  const int* __restrict__ pdf  = isDen ? dpdf : (npdf + b * A_NUM);
  const float* __restrict__ wgt = isDen ? dw : (nw + b * A_NUM);
  const float* __restrict__ initlp = isDen ? dinit : ninit;
  const float* __restrict__ finlp  = isDen ? dfin  : nfin;

  // vectorized views (den arrays: 16B-aligned base, A_DEN % 4 == 0)
  const int4*   __restrict__ src4 = (const int4*)dsrc;
  const int4*   __restrict__ dst4 = (const int4*)ddst;
  const int4*   __restrict__ pdf4 = (const int4*)dpdf;
  const float4* __restrict__ wgt4 = (const float4*)dw;
  const int A4 = A_DEN >> 2;

  const int Trun = seqlen[b];                 // alpha frozen past seqlen
  const float* __restrict__ llb = ll + (size_t)b * TMAX * NPDF;

  for (int i = tid; i < S; i += NTHR) s_alpha[i] = initlp[i];
  stage_row(llb, &s_ll[0][0], NPDF, tid, NTHR);
  wait_async0();
  __syncthreads();

  const unsigned negenc = ford(NEG_INF);

  for (int t = 0; t < Trun; ++t) {
    const int cur = t & 1, nxt = cur ^ 1;
    if (t + 1 < Trun)                          // overlap next frame's staging
      stage_row(llb + (size_t)(t + 1) * NPDF, &s_ll[nxt][0], NPDF, tid, NTHR);

    for (int i = tid; i < S; i += NTHR) { s_m[i] = negenc; s_sum[i] = 0.0f; }
    __syncthreads();

    const float* __restrict__ row = &s_ll[cur][0];

    if (isDen) {
      // pass 1: b128 arc loads, compute scores once, segment max, stash scores
      for (int i = tid; i < A4; i += NTHR) {
        if (i + NTHR < A4) {
          __builtin_prefetch((const void*)(src4 + i + NTHR), 0, 0);
          __builtin_prefetch((const void*)(dst4 + i + NTHR), 0, 0);
          __builtin_prefetch((const void*)(pdf4 + i + NTHR), 0, 0);
          __builtin_prefetch((const void*)(wgt4 + i + NTHR), 0, 0);
        }
        const int4   s4 = src4[i];
        const int4   d4 = dst4[i];
        const int4   p4 = pdf4[i];
        const float4 w4 = wgt4[i];
        float4 sc;
        sc.x = s_alpha[s4.x] + w4.x + row[p4.x];
        sc.y = s_alpha[s4.y] + w4.y + row[p4.y];
        sc.z = s_alpha[s4.z] + w4.z + row[p4.z];
        sc.w = s_alpha[s4.w] + w4.w + row[p4.w];
        atomicMax(&s_m[d4.x], ford(sc.x));
        atomicMax(&s_m[d4.y], ford(sc.y));
        atomicMax(&s_m[d4.z], ford(sc.z));
        atomicMax(&s_m[d4.w], ford(sc.w));
        ((float4*)s_score)[i] = sc;
      }
      __syncthreads();
      // pass 2: reuse stashed scores; only dst is re-read (L2-hot b128)
      for (int i = tid; i < A4; i += NTHR) {
        const int4   d4 = dst4[i];
        const float4 sc = ((const float4*)s_score)[i];
        float m, ms;
        m = funord(s_m[d4.x]); ms = (m > 0.5f * NEG_INF) ? m : 0.0f;
        atomicAdd(&s_sum[d4.x], __expf(sc.x - ms));
        m = funord(s_m[d4.y]); ms = (m > 0.5f * NEG_INF) ? m : 0.0f;
        atomicAdd(&s_sum[d4.y], __expf(sc.y - ms));
        m = funord(s_m[d4.z]); ms = (m > 0.5f * NEG_INF) ? m : 0.0f;
        atomicAdd(&s_sum[d4.z], __expf(sc.z - ms));
        m = funord(s_m[d4.w]); ms = (m > 0.5f * NEG_INF) ? m : 0.0f;
        atomicAdd(&s_sum[d4.w], __expf(sc.w - ms));
      }
    } else {
      // numerator: tiny (501 arcs), scalar two-pass with recompute
      for (int a = tid; a < A_NUM; a += NTHR) {
        float sc = s_alpha[src[a]] + wgt[a] + row[pdf[a]];
        atomicMax(&s_m[dst[a]], ford(sc));
      }
      __syncthreads();
      for (int a = tid; a < A_NUM; a += NTHR) {
        int d = dst[a];
        float sc = s_alpha[src[a]] + wgt[a] + row[pdf[a]];
        float m  = funord(s_m[d]);
        float ms = (m > 0.5f * NEG_INF) ? m : 0.0f;
        atomicAdd(&s_sum[d], __expf(sc - ms));
      }
    }
    __syncthreads();

    // alpha update
    for (int i = tid; i < S; i += NTHR) {
      float ss = s_sum[i];
      float m  = funord(s_m[i]);
      float ms = (m > 0.5f * NEG_INF) ? m : 0.0f;
      s_alpha[i] = (ss > 0.0f) ? (__logf(ss) + ms) : NEG_INF;
    }
    wait_async0();    // next frame's row is in LDS before anyone reads it
    __syncthreads();
  }

  // logsumexp(alpha + final_logp)
  float lm = NEG_INF;
  for (int i = tid; i < S; i += NTHR) lm = fmaxf(lm, s_alpha[i] + finlp[i]);
  if (tid == 0) { s_m[0] = ford(NEG_INF); s_sum[0] = 0.0f; }
  __syncthreads();
  atomicMax(&s_m[0], ford(lm));
  __syncthreads();
  float M  = funord(s_m[0]);
  float Ms = (M > 0.5f * NEG_INF) ? M : 0.0f;
  float ls = 0.0f;
  for (int i = tid; i < S; i += NTHR) ls += __expf(s_alpha[i] + finlp[i] - Ms);
  atomicAdd(&s_sum[0], ls);
  __syncthreads();
  if (tid == 0) {
    float tot = s_sum[0];
    llh_out[blockIdx.x] = (tot > 0.0f) ? (__logf(tot) + Ms) : NEG_INF;
  }
}

// ---------- final reduction: loss = sum_b (den - num), via f32 WMMA --------
typedef float v8f __attribute__((ext_vector_type(8)));
typedef float v2f __attribute__((ext_vector_type(2)));

__global__ __launch_bounds__(32) void lfmmi_loss_kernel(
    const float* __restrict__ llh, float* __restrict__ out)
{
  const int lane = threadIdx.x;                 // one wave32, EXEC all-1s
  float x = (lane < 16) ? llh[lane] : -llh[lane];
  float part;
#if __has_builtin(__builtin_amdgcn_wmma_f32_16x16x4_f32)
  // A = ones(16x4); the 32 signed llh values occupy each B register slot
  // exactly once, so sum_n D[0][n] == total sum regardless of slot mapping.
  v2f a;  a.x = 1.0f; a.y = 1.0f;
  v2f bb; bb.x = x;   bb.y = 0.0f;
  v8f c = {};
  c = __builtin_amdgcn_wmma_f32_16x16x4_f32(false, a, false, bb,
                                            (short)0, c, false, false);
  part = (lane < 16) ? c[0] : 0.0f;             // lanes 0..15 hold D[0][n]
#else
  part = x;
#endif
  __shared__ float red[32];
  red[lane] = part;
  __syncthreads();
  if (lane == 0) {
    float acc = 0.0f;
    for (int i = 0; i < 32; ++i) acc += red[i];
    out[0] = acc;
  }
}

extern "C" void kernel_launch(void* const* d_in, const int* in_sizes, int n_in,
                              void* d_out, int out_size, void* d_ws,
                              size_t ws_size, hipStream_t stream) {
  (void)in_sizes; (void)n_in; (void)out_size; (void)ws_size;
  const float* ll     = (const float*)d_in[0];
  const int*   seqlen = (const int*)d_in[1];
  const int*   nsrc   = (const int*)d_in[2];
  const int*   ndst   = (const int*)d_in[3];
  const int*   npdf   = (const int*)d_in[4];
  const float* nw     = (const float*)d_in[5];
  const float* ninit  = (const float*)d_in[6];
  const float* nfin   = (const float*)d_in[7];
  const int*   dsrc   = (const int*)d_in[8];
  const int*   ddst   = (const int*)d_in[9];
  const int*   dpdf   = (const int*)d_in[10];
  const float* dw     = (const float*)d_in[11];
  const float* dinit  = (const float*)d_in[12];
  const float* dfin   = (const float*)d_in[13];

  float* llh = (float*)d_ws;                    // 32 floats of scratch

  const size_t dyn_lds = (size_t)A_DEN * sizeof(float);  // 187.5 KB scores
  lfmmi_alpha_kernel<<<2 * BATCH, NTHR, dyn_lds, stream>>>(
      ll, seqlen, nsrc, ndst, npdf, nw, ninit, nfin,
      dsrc, ddst, dpdf, dw, dinit, dfin, llh);
  lfmmi_loss_kernel<<<1, 32, 0, stream>>>(llh, (float*)d_out);
}